// MULTIWINDOWLAYER_81286551044263
// MI455X (gfx1250) — compile-verified
//
#include <hip/hip_runtime.h>

typedef __attribute__((ext_vector_type(2))) float v2f;
typedef __attribute__((ext_vector_type(4))) float v4f;
typedef __attribute__((ext_vector_type(8))) float v8f;

#define SEQ    1024
#define DIM    256
#define NWIN   3
#define KTOT   (SEQ * NWIN * DIM)   // 786432
#define MTILES (DIM / 16)           // 16
#define KCHUNKS 48
#define WPB    8                    // waves per block (256 threads, wave32)
#define NPART  (KCHUNKS * WPB)      // 384 partial slots
#define KSUB   (KTOT / NPART)       // 2048 k-values per wave

// ---------------------------------------------------------------------------
// Kernel 1: build the concatenated pooled vector x[wi*S*D + t*D + d]
// = sum_{j in [t-w/2, t+w-w/2) ∩ [0,S)} e[j,d] / w,  w = 2<<wi
// ---------------------------------------------------------------------------
__global__ void __launch_bounds__(256)
pool_kernel(const float* __restrict__ e, float* __restrict__ x) {
    int idx = blockIdx.x * blockDim.x + threadIdx.x;   // exact grid, no tail
    int d  = idx & (DIM - 1);
    int t  = (idx >> 8) & (SEQ - 1);                   // DIM == 256
    int wi = idx / (SEQ * DIM);
    int w  = 2 << wi;                                  // 2, 4, 8
    int j0 = t - (w >> 1);
    float s = 0.0f;
    for (int j = j0; j < j0 + w; ++j)
        if (j >= 0 && j < SEQ) s += e[j * DIM + d];
    x[idx] = s * (1.0f / (float)w);
}

// ---------------------------------------------------------------------------
// Kernel 2: split-K GEMV via V_WMMA_F32_16X16X4_F32.
// A = 16 rows of W x 4 k ; B = x chunk broadcast to all 16 columns
// (every D column holds the same 16 partial dot products).
// Per lane: one b128 load of W covers two WMMAs (k-permutation trick).
// All addresses expressed as uniform-SGPR base + 32-bit element offset so the
// backend can use saddr-form global loads (no 64-bit VALU address adds in loop).
// ---------------------------------------------------------------------------
__global__ void __launch_bounds__(256)
gemv_wmma_kernel(const float* __restrict__ Wm,
                 const float* __restrict__ x,
                 float* __restrict__ part) {
    const unsigned mtile  = blockIdx.x & (MTILES - 1);
    const unsigned kchunk = blockIdx.x / MTILES;
    const unsigned wave   = threadIdx.x >> 5;
    const unsigned lane   = threadIdx.x & 31;
    const unsigned lhalf  = lane >> 4;                 // 0: K slots 0,1  1: K slots 2,3
    const unsigned mrow   = mtile * 16 + (lane & 15);
    const unsigned pslot  = kchunk * WPB + wave;       // 0..383
    const unsigned kbase  = pslot * KSUB;              // < 786432, fits u32

    // 32-bit element offsets (W is 201M elements = 805 MB < 2^31 bytes)
    const unsigned woff = mrow * (unsigned)KTOT + kbase + 4u * lhalf;
    const unsigned xoff = kbase + 4u * lhalf;

    v8f acc0 = {}, acc1 = {};
    #pragma unroll 4
    for (unsigned k = 0; k < KSUB; k += 8) {
        v4f a = __builtin_nontemporal_load((const v4f*)(Wm + (woff + k))); // stream W
        v4f b = *(const v4f*)(x + (xoff + k));                             // x: L2-hot
        v2f a01 = __builtin_shufflevector(a, a, 0, 1);
        v2f a23 = __builtin_shufflevector(a, a, 2, 3);
        v2f b01 = __builtin_shufflevector(b, b, 0, 1);
        v2f b23 = __builtin_shufflevector(b, b, 2, 3);
        // 8 args: (neg_a, A, neg_b, B, c_mod, C, reuse_a, reuse_b)
        acc0 = __builtin_amdgcn_wmma_f32_16x16x4_f32(false, a01, false, b01,
                                                     (short)0, acc0, false, false);
        acc1 = __builtin_amdgcn_wmma_f32_16x16x4_f32(false, a23, false, b23,
                                                     (short)0, acc1, false, false);
    }
    v8f acc = acc0 + acc1;

    // D layout: VGPR g, lane 0 -> (M=g, N=0), lane 16 -> (M=g+8, N=0)
    if ((lane & 15) == 0) {
        float* dst = part + pslot * DIM + mtile * 16 + lhalf * 8;
        #pragma unroll
        for (int g = 0; g < 8; ++g) dst[g] = acc[g];
    }
}

// ---------------------------------------------------------------------------
// Kernel 3: deterministic fixed-order reduction over the 384 partials + bias
// ---------------------------------------------------------------------------
__global__ void __launch_bounds__(256)
reduce_kernel(const float* __restrict__ part,
              const float* __restrict__ bias,
              float* __restrict__ out) {
    int m = threadIdx.x;
    float s = bias[m];
    for (int p = 0; p < NPART; ++p)
        s += part[p * DIM + m];
    out[m] = s;
}

extern "C" void kernel_launch(void* const* d_in, const int* in_sizes, int n_in,
                              void* d_out, int out_size, void* d_ws, size_t ws_size,
                              hipStream_t stream) {
    (void)in_sizes; (void)n_in; (void)out_size; (void)ws_size;
    const float* e    = (const float*)d_in[0];   // inputEmbeds [1024, 256]
    const float* Wm   = (const float*)d_in[1];   // W [256, 786432]
    const float* bias = (const float*)d_in[2];   // b [256]
    float* out  = (float*)d_out;                 // [256]

    float* x    = (float*)d_ws;                  // KTOT floats        (3.0 MB)
    float* part = x + KTOT;                      // NPART*DIM floats   (0.38 MB)

    pool_kernel     <<<KTOT / 256,       256, 0, stream>>>(e, x);
    gemv_wmma_kernel<<<MTILES * KCHUNKS, 256, 0, stream>>>(Wm, x, part);
    reduce_kernel   <<<1,                DIM, 0, stream>>>(part, bias, out);
}